// NeuralQuantizer_7507602833923
// MI455X (gfx1250) — compile-verified
//
#include <hip/hip_runtime.h>

#define NLEV 256
#define BLK  256
#define GRID_CAP 1024

typedef float v4f __attribute__((ext_vector_type(4)));
typedef unsigned int u32;
typedef unsigned long long u64;
typedef u32 v4u32 __attribute__((ext_vector_type(4)));
typedef int  v4i32 __attribute__((ext_vector_type(4)));
typedef int  v8i32 __attribute__((ext_vector_type(8)));

#if defined(__gfx1250__)
typedef __attribute__((address_space(1))) v4i32 gv4i;   // global int4
typedef __attribute__((address_space(3))) v4i32 lv4i;   // LDS int4
#endif

#if defined(__gfx1250__) && __has_builtin(__builtin_amdgcn_global_load_async_to_lds_b128)
#define HAVE_ASYNC_LDS 1
#endif

#if defined(__gfx1250__) && __has_builtin(__builtin_amdgcn_s_wait_asynccnt)
#define WAIT_ASYNC(n) __builtin_amdgcn_s_wait_asynccnt(n)
#elif defined(__gfx1250__)
#define WAIT_ASYNC(n) asm volatile("s_wait_asynccnt %0" :: "n"(n) : "memory")
#else
#define WAIT_ASYNC(n)
#endif

// Nearest center with reference-exact semantics:
//  - analytic guess i = round_half_down((x+1)*127.5)  (argmin picks lower idx on ties)
//  - 3-tap refinement against the real centers table (covers linspace ULP wobble)
//  - NaN x -> index 0 (matches jnp.argmin over all-NaN row)
__device__ __forceinline__ float quant1(float x, const float* sC) {
  float t = fmaf(x, 127.5f, 127.5f);
  t = fminf(fmaxf(t, 0.0f), 255.0f);          // fmaxf(NaN,0)=0 -> NaN maps to idx 0
  int i = (int)ceilf(t - 0.5f);               // round-half-down
  i = i < 0 ? 0 : (i > (NLEV - 1) ? (NLEV - 1) : i);
  int j = i;
  float best = fabsf(x - sC[i]);
  if (i > 0) {
    float dm = fabsf(x - sC[i - 1]);
    if (dm <= best) { best = dm; j = i - 1; }  // tie -> lower index
  }
  if (i < NLEV - 1) {
    float dp = fabsf(x - sC[i + 1]);
    if (dp < best) { j = i + 1; }              // strictly better only
  }
  return sC[j];
}

__device__ __forceinline__ v4f quant4(v4f xv, const float* sC) {
  v4f q;
  q.x = quant1(xv.x, sC);
  q.y = quant1(xv.y, sC);
  q.z = quant1(xv.z, sC);
  q.w = quant1(xv.w, sC);
  return q;
}

__global__ __launch_bounds__(BLK) void nq_main(const float* __restrict__ x,
                                               const float* __restrict__ centers,
                                               float* __restrict__ out, int n4) {
  __shared__ float sC[NLEV];
  __shared__ __align__(16) float tile[2][BLK * 4];
  const int tid = (int)threadIdx.x;

  // ---- stage centers into LDS via the Tensor Data Mover ----
#if defined(__gfx1250__) && __has_builtin(__builtin_amdgcn_tensor_load_to_lds) && __has_builtin(__builtin_amdgcn_s_wait_tensorcnt)
  if (tid < 32) {  // one wave issues the TDM descriptor (EXEC ignored by TDM)
    u64 ga = (u64)(size_t)centers;
    u32 lds_off = (u32)(size_t)&sC[0];  // low 32 bits of flat LDS addr == LDS byte offset
    // D# group0: count=1 | lds_addr | global_addr[56:0] | type=2 ("image")
    v4u32 g0 = { 1u,
                 lds_off,
                 (u32)ga,
                 ((u32)(ga >> 32) & 0x01FFFFFFu) | (2u << 30) };
    // D# group1: data_size=4B; tensor 256x1; tile 256x1; dim0_stride=256
    v8i32 g1 = { (int)(2u << 16),            // [17:16] data_size = 2 -> 4 bytes
                 (int)((u32)NLEV << 16),     // tensor_dim0[15:0] in bits[63:48]
                 (int)(1u << 16),            // tensor_dim0 hi=0, tensor_dim1=1
                 (int)((u32)NLEV << 16),     // tensor_dim1 hi=0, tile_dim0=256
                 1,                          // tile_dim1=1, tile_dim2=0
                 NLEV,                       // tensor_dim0_stride
                 0, 0 };
    v4i32 gz = { 0, 0, 0, 0 };
#if defined(__clang_major__) && (__clang_major__ >= 23)
    v8i32 gz8 = { 0, 0, 0, 0, 0, 0, 0, 0 };
    __builtin_amdgcn_tensor_load_to_lds(g0, g1, gz, gz, gz8, 0);
#else
    __builtin_amdgcn_tensor_load_to_lds(g0, g1, gz, gz, 0);
#endif
    __builtin_amdgcn_s_wait_tensorcnt(0);
  }
#else
  if (tid < NLEV) sC[tid] = centers[tid];
#endif
  __syncthreads();

  const int nChunks = (n4 + BLK - 1) / BLK;  // one chunk = BLK float4s = 4KB

#ifdef HAVE_ASYNC_LDS
  // Double-buffered async HBM->LDS pipeline. Each lane owns exactly its own
  // 16B slice of the tile, so no cross-wave sharing -> no barriers needed;
  // per-wave ASYNCcnt + in-order async completion give exact synchronization.
  int c = (int)blockIdx.x;
  int buf = 0;
  {
    long long i4 = (long long)c * BLK + tid;
    if (c < nChunks && i4 < n4) {
      __builtin_amdgcn_global_load_async_to_lds_b128(
          (gv4i*)(void*)(const_cast<float*>(x) + 4 * i4),
          (lv4i*)(void*)&tile[0][tid * 4], 0, 0);
    }
  }
  while (c < nChunks) {
    const int cn = c + (int)gridDim.x;
    if (cn < nChunks) {
      long long i4n = (long long)cn * BLK + tid;
      if (i4n < n4) {
        __builtin_amdgcn_global_load_async_to_lds_b128(
            (gv4i*)(void*)(const_cast<float*>(x) + 4 * i4n),
            (lv4i*)(void*)&tile[buf ^ 1][tid * 4], 0, 0);
      }
      WAIT_ASYNC(1);   // oldest (current buf) async load done; next stays in flight
    } else {
      WAIT_ASYNC(0);
    }
    asm volatile("" ::: "memory");
    long long i4 = (long long)c * BLK + tid;
    if (i4 < n4) {
      v4f xv = *(const v4f*)&tile[buf][tid * 4];         // ds_load_b128
      v4f q = quant4(xv, sC);
      __builtin_nontemporal_store(q, (v4f*)(out + 4 * i4));  // streaming store
    }
    buf ^= 1;
    c = cn;
  }
#else
  for (int c = (int)blockIdx.x; c < nChunks; c += (int)gridDim.x) {
    long long i4 = (long long)c * BLK + tid;
    if (i4 < n4) {
      v4f xv = *(const v4f*)(x + 4 * i4);
      v4f q = quant4(xv, sC);
      __builtin_nontemporal_store(q, (v4f*)(out + 4 * i4));
    }
  }
#endif
}

// Scalar tail for n % 4 != 0 (not hit for the given shapes, kept for generality).
__global__ void nq_tail(const float* __restrict__ x, const float* __restrict__ centers,
                        float* __restrict__ out, int start, int n) {
  int i = start + (int)(blockIdx.x * blockDim.x + threadIdx.x);
  if (i < n) {
    out[i] = quant1(x[i], centers);
  }
}

extern "C" void kernel_launch(void* const* d_in, const int* in_sizes, int n_in,
                              void* d_out, int out_size, void* d_ws, size_t ws_size,
                              hipStream_t stream) {
  const float* x       = (const float*)d_in[0];
  const float* centers = (const float*)d_in[1];
  float* out           = (float*)d_out;
  const int n   = in_sizes[0];     // 4*512*1024 = 2,097,152
  const int n4  = n / 4;
  const int rem = n - n4 * 4;

  if (n4 > 0) {
    int nChunks = (n4 + BLK - 1) / BLK;
    int grid = nChunks < GRID_CAP ? nChunks : GRID_CAP;
    nq_main<<<grid, BLK, 0, stream>>>(x, centers, out, n4);
  }
  if (rem > 0) {
    nq_tail<<<1, 64, 0, stream>>>(x, centers, out, n4 * 4, n);
  }
}